// RVQE_74526272520326
// MI455X (gfx1250) — compile-verified
//
#include <hip/hip_runtime.h>
#include <math.h>

// ---- problem constants (IN=3, WS=10, STAGES=1, ORDER=2) ----
#define NQb   14          // qubits
#define NST   16384       // 2^14 amplitudes
#define NPAIR 8192        // NST/2
#define NNEUR 23          // len(OUTLANES)
#define BIASF 1.5707963267948966f

typedef __attribute__((ext_vector_type(2))) float v2f;
typedef __attribute__((ext_vector_type(8))) float v8f;

// control lane for neuron with out-lane o: CTRL[o][i] = i + (i >= o), i in [0,12)
__device__ __forceinline__ int zbit(int n, int i, int o) {
    int cl = i + (i >= o);
    return (n >> (13 - cl)) & 1;
}

// =====================================================================
// Kernel 1: per-neuron rotation tables via FP32 WMMA (16x16x4)
//   One neuron per block (23 neurons x 128 blocks). W2 is staged in LDS
//   zero-padded to 12x16 so the WMMA B operand is branch-free ds_loads.
//   Q = Zb(16x12) @ W2(12x12)   (3 chained K=4 WMMAs, f32 accumulate)
//   th = BIAS + sum_j zb_j*(w1_j + Q_j);  c,s from tan(th)^4 identity
// =====================================================================
__global__ __launch_bounds__(256) void rvqe_tables(
    const float* __restrict__ w1, const float* __restrict__ w2,
    const int* __restrict__ postsel,
    float* __restrict__ Ctab, float* __restrict__ Stab,
    float* __restrict__ out_min)
{
    __shared__ float W2s[12 * 16];   // rows=K (12), cols padded to 16 with zeros
    __shared__ float W1s[16];

    if (blockIdx.x == 0 && threadIdx.x == 0)
        *out_min = postsel[0] ? __int_as_float(0x7f800000) : 1.0f;

    const int j   = blockIdx.x >> 7;        // neuron (128 blocks per neuron)
    const int blk = blockIdx.x & 127;
    const int tid = threadIdx.x;
    const int o   = (j < 20) ? (3 + (j % 10)) : (j - 20);

    if (tid < 192) {
        int row = tid >> 4, c = tid & 15;
        W2s[tid] = (c < 12) ? w2[j * 144 + row * 12 + c] : 0.0f;
    } else if (tid < 208) {
        int i = tid - 192;
        W1s[i] = (i < 12) ? w1[j * 12 + i] : 0.0f;
    }
    __syncthreads();

    const int lane = tid & 31;
    const int n0   = ((blk << 3) + (tid >> 5)) << 4;  // 16-state tile base
    const int M    = lane & 15;             // A-matrix row
    const int hi   = lane >> 4;             // half-wave select
    const int col  = lane & 15;             // B/D column

    v8f acc = {0.f, 0.f, 0.f, 0.f, 0.f, 0.f, 0.f, 0.f};
    #pragma unroll
    for (int k0 = 0; k0 < 12; k0 += 4) {
        // A 16x4 f32: VGPR0 -> K = k0 + (hi?2:0), VGPR1 -> +1
        const int ka = k0 + (hi ? 2 : 0);
        v2f a;
        a.x = (float)zbit(n0 + M, ka,     o);
        a.y = (float)zbit(n0 + M, ka + 1, o);
        // B 4x16 f32 from LDS (zero-padded): lanes 0-15 rows k0+0/1, 16-31 rows k0+2/3
        v2f bm;
        bm.x = W2s[ka * 16 + col];
        bm.y = W2s[(ka + 1) * 16 + col];
        acc = __builtin_amdgcn_wmma_f32_16x16x4_f32(
                  false, a, false, bm, (short)0, acc, false, false);
    }

    #pragma unroll
    for (int r = 0; r < 8; ++r) {
        const int n = n0 + r + (hi ? 8 : 0);   // state for this D row
        float q = acc[r];
        float val = 0.0f;
        if (col < 12) {
            float zb = (float)zbit(n, col, o);
            val = zb * (W1s[col] + q);
        }
        // reduce the 16 columns within each half-wave
        val += __shfl_xor(val, 1, 32);
        val += __shfl_xor(val, 2, 32);
        val += __shfl_xor(val, 4, 32);
        val += __shfl_xor(val, 8, 32);

        const float th = BIASF + val;
        const float t  = tanf(th);
        const float t2 = t * t;
        const float x  = t2 * t2;              // tan(th)^4 >= 0
        float c, s;
        if (x > 1e18f) { c = 1.0f / x; s = 1.0f; }
        else {
            float inv = 1.0f / sqrtf(fmaf(x, x, 1.0f));
            c = inv; s = x * inv;
        }
        if (col == r) {                         // lanes r and 16+r write
            Ctab[j * NST + n] = c;
            Stab[j * NST + n] = s;
        }
    }
}

// =====================================================================
// Kernel 2: full 16-step scan, state resident in LDS (64 KB / 320 KB)
// =====================================================================
__device__ __forceinline__ void bitflip_lds(float* st, int mask, int tid) {
    if (mask) {
        const int msb = 1 << (31 - __clz(mask));
        const int low = msb - 1;
        for (int k = tid; k < NPAIR; k += 1024) {
            int n  = ((k & ~low) << 1) | (k & low);   // insert 0 at msb position
            int n2 = n ^ mask;
            float tmp = st[n]; st[n] = st[n2]; st[n2] = tmp;
        }
    }
    __syncthreads();
}

__device__ __forceinline__ void rot_tab(float* st,
                                        const float* __restrict__ C,
                                        const float* __restrict__ S,
                                        int o, int tid) {
    const int logR = 13 - o;
    const int Rm   = (1 << logR) - 1;
    for (int pi = tid; pi < NPAIR; pi += 1024) {
        int r  = pi & Rm, l = pi >> logR;
        int i0 = (l << (logR + 1)) | r;
        int i1 = i0 | (Rm + 1);
        float c = C[i0], s = S[i0];
        float a0 = st[i0], a1 = st[i1];
        st[i0] = c * a0 - s * a1;
        st[i1] = s * a0 + c * a1;
    }
}

__device__ __forceinline__ void rot_sc(float* st, float c, float s, int o, int tid) {
    const int logR = 13 - o;
    const int Rm   = (1 << logR) - 1;
    for (int pi = tid; pi < NPAIR; pi += 1024) {
        int r  = pi & Rm, l = pi >> logR;
        int i0 = (l << (logR + 1)) | r;
        int i1 = i0 | (Rm + 1);
        float a0 = st[i0], a1 = st[i1];
        st[i0] = c * a0 - s * a1;
        st[i1] = s * a0 + c * a1;
    }
}

// prefetch an upcoming 64KB table pass into cache (gfx1250 global_prefetch_b8)
__device__ __forceinline__ void pf_tab(const float* __restrict__ C,
                                       const float* __restrict__ S, int tid) {
    __builtin_prefetch(C + (tid << 4), 0, 0);
    __builtin_prefetch(S + (tid << 4), 0, 0);
}

__global__ __launch_bounds__(1024) void rvqe_scan(
    const int* __restrict__ inputs, const int* __restrict__ targets,
    const float* __restrict__ ua, const int* __restrict__ postsel,
    const float* __restrict__ Ctab, const float* __restrict__ Stab,
    float* __restrict__ out_probs, float* __restrict__ out_meas,
    float* __restrict__ out_min)
{
    __shared__ float st[NST];
    __shared__ float red8[32 * 8];
    __shared__ float pbuf[8];

    const int tid = threadIdx.x;
    const int b   = blockIdx.x;
    const int PS  = postsel[0];
    const int wid = tid >> 5, lane = tid & 31;

    float cu[10], su[10];
    #pragma unroll
    for (int i = 0; i < 10; ++i) {
        float h = 0.5f * ua[i];
        cu[i] = cosf(h); su[i] = sinf(h);
    }

    for (int n = tid; n < NST; n += 1024) st[n] = 0.0f;
    if (tid == 0) st[0] = 1.0f;
    __syncthreads();

    for (int t = 0; t < 16; ++t) {
        int i_int = 0, t_int = 0;
        #pragma unroll
        for (int i = 0; i < 3; ++i) {
            i_int = (i_int << 1) | inputs [(b * 17 + t)     * 3 + i];
            t_int = (t_int << 1) | targets[(b * 17 + t + 1) * 3 + i];
        }

        bitflip_lds(st, i_int << 11, tid);
        for (int k = 0; k < 10; ++k) {
            if (k < 9)
                pf_tab(Ctab + (size_t)(k + 1) * NST, Stab + (size_t)(k + 1) * NST, tid);
            rot_tab(st, Ctab + (size_t)k * NST, Stab + (size_t)k * NST, 3 + k, tid);
            __syncthreads();
        }
        for (int li = 0; li < 10; ++li) {
            if (li == 0)
                pf_tab(Ctab + (size_t)10 * NST, Stab + (size_t)10 * NST, tid);
            rot_sc(st, cu[li], su[li], 3 + li, tid);
            __syncthreads();
        }
        for (int k = 10; k < 20; ++k) {
            if (k < 19)
                pf_tab(Ctab + (size_t)(k + 1) * NST, Stab + (size_t)(k + 1) * NST, tid);
            rot_tab(st, Ctab + (size_t)k * NST, Stab + (size_t)k * NST, 3 + (k - 10), tid);
            __syncthreads();
        }
        bitflip_lds(st, i_int << 11, tid);
        for (int k = 20; k < 23; ++k) {
            if (k < 22)
                pf_tab(Ctab + (size_t)(k + 1) * NST, Stab + (size_t)(k + 1) * NST, tid);
            rot_tab(st, Ctab + (size_t)k * NST, Stab + (size_t)k * NST, k - 20, tid);
            __syncthreads();
        }

        // p[m] = sum of st^2 over contiguous 2048-chunk m  (2 barriers total)
        {
            float part[8];
            #pragma unroll
            for (int m = 0; m < 8; ++m) {
                float a  = st[(m << 11) + tid];
                float b2 = st[(m << 11) + 1024 + tid];
                part[m] = a * a + b2 * b2;
                for (int off = 16; off; off >>= 1)
                    part[m] += __shfl_down(part[m], off, 32);
            }
            if (lane == 0) {
                #pragma unroll
                for (int m = 0; m < 8; ++m) red8[wid * 8 + m] = part[m];
            }
            __syncthreads();
            if (wid < 8) {                       // wave w reduces outcome m=w
                float x = red8[lane * 8 + wid];
                for (int off = 16; off; off >>= 1) x += __shfl_down(x, off, 32);
                if (lane == 0) pbuf[wid] = x;
            }
            __syncthreads();
        }

        int m_int;
        if (PS) m_int = t_int;
        else {
            float best = pbuf[0]; int bi = 0;
            for (int m = 1; m < 8; ++m) if (pbuf[m] > best) { best = pbuf[m]; bi = m; }
            m_int = bi;
        }
        const float p_sel = pbuf[m_int];
        const float inv   = 1.0f / sqrtf(p_sel + 1e-12f);

        // collapse+normalize+bitflip fused: new[n] = (top(n)==0) ? st[n^mmask]*inv : 0
        const int mmask = m_int << 11;
        float v0 = st[mmask | tid]          * inv;
        float v1 = st[mmask | (1024 + tid)] * inv;
        __syncthreads();
        for (int n = tid; n < NST; n += 1024) st[n] = 0.0f;
        st[tid] = v0; st[tid + 1024] = v1;
        __syncthreads();

        if (tid < 8) out_probs[(b * 8 + tid) * 16 + t] = pbuf[tid];
        if (tid == 0) {
            out_meas[(b * 16 + t) * 3 + 0] = (float)((m_int >> 2) & 1);
            out_meas[(b * 16 + t) * 3 + 1] = (float)((m_int >> 1) & 1);
            out_meas[(b * 16 + t) * 3 + 2] = (float)(m_int & 1);
            if (PS) atomicMin((int*)out_min, __float_as_int(p_sel));
        }
    }
}

extern "C" void kernel_launch(void* const* d_in, const int* in_sizes, int n_in,
                              void* d_out, int out_size, void* d_ws, size_t ws_size,
                              hipStream_t stream) {
    const int*   inputs  = (const int*)  d_in[0];   // (32,17,3) int32
    const int*   targets = (const int*)  d_in[1];   // (32,17,3) int32
    const float* w1      = (const float*)d_in[2];   // (23,12)
    const float* w2      = (const float*)d_in[3];   // (23,12,12)
    const float* ua      = (const float*)d_in[4];   // (1,10)
    const int*   postsel = (const int*)  d_in[5];   // scalar

    float* out       = (float*)d_out;
    float* out_probs = out;                 // 32*8*16 = 4096
    float* out_meas  = out + 4096;          // 32*16*3 = 1536
    float* out_min   = out + 5632;          // 1

    float* Ctab = (float*)d_ws;             // 23*16384 floats
    float* Stab = Ctab + NNEUR * NST;       // 23*16384 floats (total ~3 MB)

    rvqe_tables<<<NNEUR * 128, 256, 0, stream>>>(w1, w2, postsel, Ctab, Stab, out_min);
    rvqe_scan<<<32, 1024, 0, stream>>>(inputs, targets, ua, postsel,
                                       Ctab, Stab, out_probs, out_meas, out_min);
}